// ClientGCN_86912958202587
// MI455X (gfx1250) — compile-verified
//
#include <hip/hip_runtime.h>
#include <hip/hip_bf16.h>

#define NV 6144
#define FD 512
constexpr float LEAKY = 0.2f;
constexpr float EPSV  = 1e-12f;

typedef __attribute__((ext_vector_type(2))) float v2f;
typedef __attribute__((ext_vector_type(8))) float v8f;

// ---------------------------------------------------------------------------
// K1: Wh = X @ W via V_WMMA_F32_16X16X4_F32, then wh1 = Wh@a[:16], wh2 = Wh@a[16:]
// One wave per 16-row tile of X; 8 waves / block; W staged in LDS (32 KB).
// A layout (16x4 f32): lanes 0-15 row M=lane,   v0=K0,v1=K1
//                      lanes16-31 row M=lane-16, v0=K2,v1=K3
// B layout (4x16 f32): lanes 0-15 col N=lane,   v0=K0,v1=K1
//                      lanes16-31 col N=lane-16, v0=K2,v1=K3
// D layout (16x16 f32): VGPR r: lanes0-15 -> (M=r, N=lane); lanes16-31 -> (M=r+8).
// ---------------------------------------------------------------------------
__global__ __launch_bounds__(256) void k_gemm_wmma(
    const float* __restrict__ X, const float* __restrict__ W,
    const float* __restrict__ A, float* __restrict__ wh1, float* __restrict__ wh2) {
  __shared__ float Wl[FD * 16];
  for (int t = threadIdx.x; t < FD * 16 / 4; t += 256)
    ((float4*)Wl)[t] = ((const float4*)W)[t];
  __syncthreads();

  const int wave = threadIdx.x >> 5, lane = threadIdx.x & 31;
  const int half = lane >> 4, l16 = lane & 15;
  const int tile = blockIdx.x * 8 + wave;
  const int row0 = tile * 16;
  const float* xrow = X + (size_t)(row0 + l16) * FD;

  v8f c = {0.f, 0.f, 0.f, 0.f, 0.f, 0.f, 0.f, 0.f};
  for (int k = 0; k < FD; k += 4) {
    const int ka = k + half * 2;
    v2f av; av.x = xrow[ka];            av.y = xrow[ka + 1];
    v2f bv; bv.x = Wl[ka * 16 + l16];   bv.y = Wl[(ka + 1) * 16 + l16];
    c = __builtin_amdgcn_wmma_f32_16x16x4_f32(false, av, false, bv,
                                              (short)0, c, false, false);
  }

  // Fold 16-wide dot products with a[:16] / a[16:32] across each 16-lane half.
  const float a1 = A[l16], a2 = A[16 + l16];
  float s1[8], s2[8];
#pragma unroll
  for (int r = 0; r < 8; ++r) { s1[r] = c[r] * a1; s2[r] = c[r] * a2; }
#pragma unroll
  for (int off = 8; off >= 1; off >>= 1) {
#pragma unroll
    for (int r = 0; r < 8; ++r) {
      s1[r] += __shfl_xor(s1[r], off, 32);
      s2[r] += __shfl_xor(s2[r], off, 32);
    }
  }
  if (l16 == 0) {
#pragma unroll
    for (int r = 0; r < 8; ++r) {
      wh1[row0 + half * 8 + r] = s1[r];
      wh2[row0 + half * 8 + r] = s2[r];
    }
  }
}

// ---------------------------------------------------------------------------
// K2: per-row masked-softmax stats (online max + sumexp over adj>0 entries).
// Masked entries contribute exp(-9e15 - max) == 0 in fp32, so skipping them
// is bit-exact. One 256-thread block per row.
// ---------------------------------------------------------------------------
__global__ __launch_bounds__(256) void k_row_stats(
    const float* __restrict__ adj, const float* __restrict__ wh1,
    const float* __restrict__ wh2, float* __restrict__ rowmax,
    float* __restrict__ rowis) {
  const int i = blockIdx.x;
  const int tid = threadIdx.x;
  const float w1 = wh1[i];
  float m = -3.0e38f, s = 0.f;
  for (int j = tid; j < NV; j += 256) {
    if (adj[(size_t)i * NV + j] > 0.f) {
      float t = w1 + wh2[j];
      float ev = t > 0.f ? t : LEAKY * t;
      float nm = fmaxf(m, ev);
      s = s * __expf(m - nm) + __expf(ev - nm);
      m = nm;
    }
  }
  __shared__ float ms[256], ss[256];
  ms[tid] = m; ss[tid] = s;
  __syncthreads();
  for (int off = 128; off > 0; off >>= 1) {
    if (tid < off) {
      float m2 = ms[tid + off], s2 = ss[tid + off];
      float nm = fmaxf(m, m2);
      s = s * __expf(m - nm) + s2 * __expf(m2 - nm);
      m = nm;
      ms[tid] = m; ss[tid] = s;
    }
    __syncthreads();
  }
  if (tid == 0) { rowmax[i] = m; rowis[i] = 1.f / s; }
}

__global__ void k_zero(float* __restrict__ p, int n) {
  int i = blockIdx.x * 256 + threadIdx.x;
  if (i < n) p[i] = 0.f;
}

// ---------------------------------------------------------------------------
// K3: 64x64 tiles over (i,j). All tiles write e. Tiles on/above the diagonal
// compute hard = round(sigmoid(2*(logit(p)+logistic_noise))) for j>i, stage
// in LDS, write tile + transposed mirror coalesced, and accumulate exact
// 0/1 degree sums via atomics (deterministic: integer-valued floats).
// ---------------------------------------------------------------------------
__global__ __launch_bounds__(256) void k_main_tile(
    const float* __restrict__ adj, const float* __restrict__ u,
    const float* __restrict__ wh1, const float* __restrict__ wh2,
    const float* __restrict__ rowmax, const float* __restrict__ rowis,
    float* __restrict__ out_adj, float* __restrict__ out_e,
    float* __restrict__ deg) {
  const int ti = blockIdx.y, tj = blockIdx.x;
  const int i0 = ti * 64, j0 = tj * 64;
  const int tid = threadIdx.x;
  const int tx = tid & 63, ty = tid >> 6;

  __shared__ float h[64][65];
  __shared__ float sw1[64], srm[64], sri[64];
  if (tid < 64) {
    sw1[tid] = wh1[i0 + tid];
    srm[tid] = rowmax[i0 + tid];
    sri[tid] = rowis[i0 + tid];
  }
  __syncthreads();

  const bool diag = (ti == tj), upper = (tj > ti);
  const int jj = j0 + tx;
  const float w2 = wh2[jj];

  for (int rr = ty; rr < 64; rr += 4) {
    const int ii = i0 + rr;
    const float t = sw1[rr] + w2;
    const float ev = t > 0.f ? t : LEAKY * t;
    out_e[(size_t)ii * NV + jj] = ev;
    if (upper || diag) {
      float hard = 0.f;
      if (!(diag && jj <= ii)) {               // strict upper triangle only
        const float av = adj[(size_t)ii * NV + jj];
        const float aorig = av > 0.f ? 1.f : 0.f;
        float att = 0.f;
        if (av > 0.f) att = __expf(ev - srm[rr]) * sri[rr];
        const float p  = 0.5f * att + 0.5f * aorig;
        const float lg = __logf(p + EPSV) - log1pf(EPSV - p);
        const float uv = u[(size_t)ii * NV + jj];
        const float nz = __logf(uv) - log1pf(-uv);
        const float soft = 1.f / (1.f + __expf(-2.f * (lg + nz))); // temp=0.5
        hard = rintf(soft);                    // round-half-even, matches jnp
      }
      h[rr][tx] = hard;
    }
  }
  __syncthreads();

  if (upper) {
    for (int rr = ty; rr < 64; rr += 4) {
      out_adj[(size_t)(i0 + rr) * NV + j0 + tx] = h[rr][tx];
      out_adj[(size_t)(j0 + rr) * NV + i0 + tx] = h[tx][rr];  // mirror
    }
    if (ty == 0) {
      float rs = 0.f;
      for (int c2 = 0; c2 < 64; ++c2) rs += h[tx][c2];
      atomicAdd(&deg[i0 + tx], rs);
    } else if (ty == 1) {
      float cs = 0.f;
      for (int r2 = 0; r2 < 64; ++r2) cs += h[r2][tx];
      atomicAdd(&deg[j0 + tx], cs);
    }
  } else if (diag) {
    for (int rr = ty; rr < 64; rr += 4) {
      float v = (tx == rr) ? 1.f : (tx > rr ? h[rr][tx] : h[tx][rr]);
      out_adj[(size_t)(i0 + rr) * NV + j0 + tx] = v;
    }
    if (ty == 0) {
      float rs = 0.f;
      for (int c2 = 0; c2 < 64; ++c2) rs += h[tx][c2];
      atomicAdd(&deg[i0 + tx], rs);
    } else if (ty == 1) {
      float cs = 0.f;
      for (int r2 = 0; r2 < 64; ++r2) cs += h[r2][tx];
      atomicAdd(&deg[j0 + tx], cs);
    }
  }
}

__global__ void k_dvec(const float* __restrict__ deg, float* __restrict__ dv) {
  int i = blockIdx.x * 256 + threadIdx.x;
  if (i < NV) dv[i] = rsqrtf(deg[i] + 1.0f);   // +1 for unit diagonal
}

// K5: in-place adj_norm = d_i * s * d_j, float4-vectorized.
__global__ __launch_bounds__(256) void k_scale(float* __restrict__ out_adj,
                                               const float* __restrict__ dv) {
  const int i  = blockIdx.y;
  const int j4 = blockIdx.x * 256 + threadIdx.x;       // float4 index in row
  const float di = dv[i];
  float4 v  = ((float4*)out_adj)[(size_t)i * (NV / 4) + j4];
  const float4 dj = ((const float4*)dv)[j4];
  v.x *= di * dj.x; v.y *= di * dj.y; v.z *= di * dj.z; v.w *= di * dj.w;
  ((float4*)out_adj)[(size_t)i * (NV / 4) + j4] = v;
}

extern "C" void kernel_launch(void* const* d_in, const int* in_sizes, int n_in,
                              void* d_out, int out_size, void* d_ws, size_t ws_size,
                              hipStream_t stream) {
  (void)in_sizes; (void)n_in; (void)out_size; (void)ws_size;
  const float* X   = (const float*)d_in[0];
  const float* W   = (const float*)d_in[1];
  const float* A   = (const float*)d_in[2];
  const float* adj = (const float*)d_in[3];
  const float* u   = (const float*)d_in[4];

  float* out     = (float*)d_out;
  float* out_adj = out;                         // adj_norm first (tuple order)
  float* out_e   = out + (size_t)NV * NV;       // e second

  float* ws     = (float*)d_ws;                 // 6*NV floats (~147 KB)
  float* wh1    = ws;
  float* wh2    = ws + NV;
  float* rowmax = ws + 2 * NV;
  float* rowis  = ws + 3 * NV;
  float* deg    = ws + 4 * NV;
  float* dv     = ws + 5 * NV;

  k_gemm_wmma<<<NV / 128, 256, 0, stream>>>(X, W, A, wh1, wh2);
  k_zero<<<(NV + 255) / 256, 256, 0, stream>>>(deg, NV);
  k_row_stats<<<NV, 256, 0, stream>>>(adj, wh1, wh2, rowmax, rowis);
  dim3 g3(NV / 64, NV / 64);
  k_main_tile<<<g3, 256, 0, stream>>>(adj, u, wh1, wh2, rowmax, rowis,
                                      out_adj, out_e, deg);
  k_dvec<<<(NV + 255) / 256, 256, 0, stream>>>(deg, dv);
  dim3 g5(NV / 1024, NV);
  k_scale<<<g5, 256, 0, stream>>>(out_adj, dv);
}